// BilateralSliceApply_81982335746351
// MI455X (gfx1250) — compile-verified
//
#include <hip/hip_runtime.h>
#include <hip/hip_bf16.h>

typedef float v4f __attribute__((ext_vector_type(4)));

#define B_      4
#define CIN     3
#define CGRID   12
#define DZ      8
#define HG      16
#define WG      16
#define H_      1024
#define W_      1024
#define GRID_PER_B (CGRID * DZ * HG * WG)   // 24576 floats = 96 KB
#define ROWS_PER_BLOCK 8
#define THREADS 256

// ---- CDNA5 async global->LDS path (guarded; falls back to plain copy) ----
#if defined(__gfx1250__) && __has_builtin(__builtin_amdgcn_global_load_async_to_lds_b32)
#define USE_ASYNC_LDS 1
#else
#define USE_ASYNC_LDS 0
#endif

#if USE_ASYNC_LDS
typedef __attribute__((address_space(1))) int* gas_i32p;
typedef __attribute__((address_space(3))) int* las_i32p;
#endif

__device__ __forceinline__ void wait_async_zero() {
#if defined(__gfx1250__) && __has_builtin(__builtin_amdgcn_s_wait_asynccnt)
    __builtin_amdgcn_s_wait_asynccnt(0);
#elif USE_ASYNC_LDS
    asm volatile("s_wait_asynccnt 0" ::: "memory");
#endif
}

__global__ __launch_bounds__(THREADS)
void bilateral_slice_apply_kernel(const float* __restrict__ grid,
                                  const float* __restrict__ guide,
                                  const float* __restrict__ image,
                                  float* __restrict__ out) {
    // Transposed grid for this batch: [z][y][x][c] (c contiguous -> 3x b128/corner)
    __shared__ float sgrid[GRID_PER_B];

    const int b   = blockIdx.z;
    const int tid = threadIdx.x;

    // ---- Stage + transpose this batch's bilateral grid into LDS (async DMA) ----
    const float* gbase = grid + (size_t)b * GRID_PER_B;   // layout [c][z][y][x]
    for (int i = tid; i < GRID_PER_B; i += THREADS) {
        const int c   = i >> 11;           // / (8*16*16)
        const int rem = i & 2047;
        const int z   = rem >> 8;
        const int y   = (rem >> 4) & 15;
        const int x   = rem & 15;
        const int dst = ((z * HG + y) * WG + x) * CGRID + c;
#if USE_ASYNC_LDS
        __builtin_amdgcn_global_load_async_to_lds_b32(
            (gas_i32p)(gbase + i),
            (las_i32p)(&sgrid[dst]),
            0, 0);
#else
        sgrid[dst] = gbase[i];
#endif
    }
    wait_async_zero();
    __syncthreads();

    const int   row0   = blockIdx.y * ROWS_PER_BLOCK;
    const int   x0pix  = tid * 4;                 // 4 consecutive pixels / thread
    const float inv64  = 1.0f / 64.0f;            // 16/1024

    for (int r = 0; r < ROWS_PER_BLOCK; ++r) {
        const int row = row0 + r;

        // y interpolation (uniform across the row -> scalar regs)
        const float gy  = (row + 0.5f) * inv64;
        const float fyf = floorf(gy - 0.5f);
        const float wy1f = gy - 0.5f - fyf;
        const float wy0f = 1.0f - wy1f;
        const int   yib  = (int)fyf;
        const int   yi0  = min(max(yib, 0), HG - 1);
        const int   yi1  = min(max(yib + 1, 0), HG - 1);

        // streaming loads (non-temporal: zero reuse)
        const size_t prow = ((size_t)b * H_ + row) * W_ + x0pix;
        const v4f gd  = __builtin_nontemporal_load((const v4f*)(guide + prow));
        const size_t irow = (((size_t)b * CIN) * H_ + row) * W_ + x0pix;
        const v4f im0 = __builtin_nontemporal_load((const v4f*)(image + irow));
        const v4f im1 = __builtin_nontemporal_load((const v4f*)(image + irow + (size_t)H_ * W_));
        const v4f im2 = __builtin_nontemporal_load((const v4f*)(image + irow + 2 * (size_t)H_ * W_));

        // prefetch next row's streams while we compute
        if (r + 1 < ROWS_PER_BLOCK) {
            __builtin_prefetch(guide + prow + W_, 0, 0);
            __builtin_prefetch(image + irow + W_, 0, 0);
            __builtin_prefetch(image + irow + (size_t)H_ * W_ + W_, 0, 0);
            __builtin_prefetch(image + irow + 2 * (size_t)H_ * W_ + W_, 0, 0);
        }

        v4f o0, o1, o2;

        #pragma unroll
        for (int p = 0; p < 4; ++p) {
            const int   xpix = x0pix + p;
            const float gx   = (xpix + 0.5f) * inv64;
            const float fxf  = floorf(gx - 0.5f);
            const float wx1f = gx - 0.5f - fxf;
            const float wx0f = 1.0f - wx1f;
            const int   xib  = (int)fxf;
            const int   xi0  = min(max(xib, 0), WG - 1);
            const int   xi1  = min(max(xib + 1, 0), WG - 1);

            const float gz   = gd[p] * (float)DZ;
            const float fzf  = floorf(gz - 0.5f);
            const float wz1f = gz - 0.5f - fzf;
            const float wz0f = 1.0f - wz1f;
            const int   zib  = (int)fzf;
            const int   zi0  = min(max(zib, 0), DZ - 1);
            const int   zi1  = min(max(zib + 1, 0), DZ - 1);

            const int   zi[2] = {zi0, zi1};  const float wz[2] = {wz0f, wz1f};
            const int   yi[2] = {yi0, yi1};  const float wy[2] = {wy0f, wy1f};
            const int   xi[2] = {xi0, xi1};  const float wx[2] = {wx0f, wx1f};

            float acc[12];
            #pragma unroll
            for (int k = 0; k < 12; ++k) acc[k] = 0.0f;

            // 8 trilinear corners; 3x ds_load_b128 each (48B-aligned bases)
            #pragma unroll
            for (int dz = 0; dz < 2; ++dz) {
                #pragma unroll
                for (int dy = 0; dy < 2; ++dy) {
                    const float wzy = wz[dz] * wy[dy];
                    const int   bzy = (zi[dz] * HG + yi[dy]) * WG;
                    #pragma unroll
                    for (int dx = 0; dx < 2; ++dx) {
                        const float  w  = wzy * wx[dx];
                        const float* cp = &sgrid[(bzy + xi[dx]) * CGRID];
                        const v4f q0 = *(const v4f*)(cp);
                        const v4f q1 = *(const v4f*)(cp + 4);
                        const v4f q2 = *(const v4f*)(cp + 8);
                        acc[0]  = fmaf(w, q0[0], acc[0]);
                        acc[1]  = fmaf(w, q0[1], acc[1]);
                        acc[2]  = fmaf(w, q0[2], acc[2]);
                        acc[3]  = fmaf(w, q0[3], acc[3]);
                        acc[4]  = fmaf(w, q1[0], acc[4]);
                        acc[5]  = fmaf(w, q1[1], acc[5]);
                        acc[6]  = fmaf(w, q1[2], acc[6]);
                        acc[7]  = fmaf(w, q1[3], acc[7]);
                        acc[8]  = fmaf(w, q2[0], acc[8]);
                        acc[9]  = fmaf(w, q2[1], acc[9]);
                        acc[10] = fmaf(w, q2[2], acc[10]);
                        acc[11] = fmaf(w, q2[3], acc[11]);
                    }
                }
            }

            // apply: out[c] = A[c][0..2] . rgb + A[c][3]
            o0[p] = fmaf(acc[0], im0[p], fmaf(acc[1],  im1[p], fmaf(acc[2],  im2[p], acc[3])));
            o1[p] = fmaf(acc[4], im0[p], fmaf(acc[5],  im1[p], fmaf(acc[6],  im2[p], acc[7])));
            o2[p] = fmaf(acc[8], im0[p], fmaf(acc[9],  im1[p], fmaf(acc[10], im2[p], acc[11])));
        }

        float* obase = out + (((size_t)b * CIN) * H_ + row) * W_ + x0pix;
        __builtin_nontemporal_store(o0, (v4f*)(obase));
        __builtin_nontemporal_store(o1, (v4f*)(obase + (size_t)H_ * W_));
        __builtin_nontemporal_store(o2, (v4f*)(obase + 2 * (size_t)H_ * W_));
    }
}

extern "C" void kernel_launch(void* const* d_in, const int* in_sizes, int n_in,
                              void* d_out, int out_size, void* d_ws, size_t ws_size,
                              hipStream_t stream) {
    const float* grid  = (const float*)d_in[0];   // [4,12,8,16,16]
    const float* guide = (const float*)d_in[1];   // [4,1024,1024]
    const float* image = (const float*)d_in[2];   // [4,3,1024,1024]
    float*       out   = (float*)d_out;           // [4,3,1024,1024]

    dim3 blocks(1, H_ / ROWS_PER_BLOCK, B_);      // 1 x 128 x 4 = 512 blocks
    bilateral_slice_apply_kernel<<<blocks, dim3(THREADS), 0, stream>>>(grid, guide, image, out);
}